// ChamfersLoss_36996848287806
// MI455X (gfx1250) — compile-verified
//
#include <hip/hip_runtime.h>
#include <math.h>

// CDNA5 (gfx1250) wave32 WMMA fragments for V_WMMA_F32_16X16X4_F32:
//   A: 16x4 f32  -> 2 VGPRs/lane (v2f)   B: 4x16 f32 -> 2 VGPRs/lane (v2f)
//   C/D: 16x16 f32 -> 8 VGPRs/lane (v8f)
typedef __attribute__((ext_vector_type(2))) float v2f;
typedef __attribute__((ext_vector_type(8))) float v8f;

#define WAVE 32
#define BLOCK 128  // 4 waves per workgroup

// Pre-pass: pack each point [3 floats] into aligned [q0, q1, q2, |q|^2].
// Lets the min-pass fetch a B-fragment half with a single aligned b64 load.
__global__ __launch_bounds__(256) void pack_points(
    const float* __restrict__ src, float* __restrict__ dst, int total)
{
    const int i = blockIdx.x * 256 + threadIdx.x;
    if (i >= total) return;
    const float q0 = src[3 * i + 0];
    const float q1 = src[3 * i + 1];
    const float q2 = src[3 * i + 2];
    const float n2 = fmaf(q2, q2, fmaf(q1, q1, q0 * q0));
    float4 v = make_float4(q0, q1, q2, n2);
    *(float4*)(dst + 4 * (size_t)i) = v;
}

// One wave owns a 16-row tile of P ([nP,3] raw) and scans all of Q (packed
// [nQ,4]) in 16-column tiles. Trick: A row = (-2p0,-2p1,-2p2, 1),
// B col = (q0,q1,q2,|q|^2)  =>  one WMMA gives acc = -2 p.q + |q|^2.
// Track min(acc) over all columns; add |p|^2 once at the end:
//   min_q d^2 = |p|^2 + min_q(acc).
__global__ __launch_bounds__(BLOCK) void chamfer_min_pass(
    const float* __restrict__ P, const float* __restrict__ Qpk,
    float* __restrict__ outMin, int nP, int nQ)
{
    const int tid  = blockIdx.x * BLOCK + threadIdx.x;
    const int wave = tid >> 5;
    const int lane = threadIdx.x & (WAVE - 1);
    const int half = lane >> 4;   // 0: lanes 0-15 (K=0,1), 1: lanes 16-31 (K=2,3)
    const int lidx = lane & 15;

    const int tilesP  = nP >> 4;
    const int batch   = wave / tilesP;
    const int tileRow = wave % tilesP;

    const float* Pb  = P   + (size_t)batch * nP * 3;
    const float* Qpb = Qpk + (size_t)batch * nQ * 4;

    // --- A fragment (fixed for the whole sweep) ---
    const int m = (tileRow << 4) + lidx;
    const float px0 = Pb[3 * m + 0];
    const float px1 = Pb[3 * m + 1];
    const float px2 = Pb[3 * m + 2];
    v2f afrag;
    afrag.x = half ? (-2.0f * px2) : (-2.0f * px0);  // K=0 / K=2
    afrag.y = half ? 1.0f          : (-2.0f * px1);  // K=1 / K=3 (=1: picks |q|^2)

    const float x2full = fmaf(px2, px2, fmaf(px1, px1, px0 * px0));
    // |p|^2 for the 8 rows this lane's C registers cover (M = half*8 + r).
    float x2row[8];
#pragma unroll
    for (int r = 0; r < 8; ++r)
        x2row[r] = __shfl(x2full, half * 8 + r, WAVE);

    float rm[8];
#pragma unroll
    for (int r = 0; r < 8; ++r) rm[r] = 3.402823466e+38f;

    // Per-lane pointer to its 8-byte B-fragment slice: col lidx, half offset.
    const float* qptr = Qpb + 4 * (size_t)lidx + 2 * half;
    const int tilesQ = nQ >> 4;
#pragma unroll 4
    for (int ct = 0; ct < tilesQ; ++ct) {
        const v2f bfrag = *(const v2f*)qptr;  // aligned b64 load
        qptr += 64;                           // 16 cols * 4 floats

        v8f acc = {};
        acc = __builtin_amdgcn_wmma_f32_16x16x4_f32(
            /*neg_a=*/false, afrag, /*neg_b=*/false, bfrag,
            /*c_mod=*/(short)0, acc, /*reuse_a=*/false, /*reuse_b=*/false);

#pragma unroll
        for (int r = 0; r < 8; ++r)
            rm[r] = fminf(rm[r], acc[r]);
    }

    // Min across the 16 lanes of each half (masks < 16 keep halves separate).
#pragma unroll
    for (int s = 8; s >= 1; s >>= 1) {
#pragma unroll
        for (int r = 0; r < 8; ++r)
            rm[r] = fminf(rm[r], __shfl_xor(rm[r], s, WAVE));
    }

    // Lane 0 holds rows M=0..7 of the tile, lane 16 holds M=8..15.
    if (lidx == 0) {
        float* dst = outMin + (size_t)batch * nP + (tileRow << 4) + half * 8;
#pragma unroll
        for (int r = 0; r < 8; ++r) dst[r] = x2row[r] + rm[r];  // min d^2
    }
}

// Deterministic single-workgroup sum of sqrt(max(d2,0)) over n partial mins.
__global__ __launch_bounds__(256) void chamfer_reduce(
    const float* __restrict__ vals, int n, float* __restrict__ out)
{
    __shared__ float sm[256];
    const int tid = threadIdx.x;
    float s = 0.0f;
    for (int i = tid; i < n; i += 256)
        s += sqrtf(fmaxf(vals[i], 0.0f));
    sm[tid] = s;
    __syncthreads();
    for (int w = 128; w >= 1; w >>= 1) {
        if (tid < w) sm[tid] += sm[tid + w];
        __syncthreads();
    }
    if (tid == 0) out[0] = sm[0];
}

extern "C" void kernel_launch(void* const* d_in, const int* in_sizes, int n_in,
                              void* d_out, int out_size, void* d_ws, size_t ws_size,
                              hipStream_t stream) {
    (void)n_in; (void)out_size; (void)ws_size;
    const float* x = (const float*)d_in[0];  // [B, N, 3]
    const float* y = (const float*)d_in[1];  // [B, M, 3]
    float* out = (float*)d_out;

    const int B = 4;
    const int N = in_sizes[0] / (B * 3);
    const int M = in_sizes[1] / (B * 3);

    // Workspace layout (floats): pkX | pkY | rowMin | colMin
    float* pkX    = (float*)d_ws;                   // B*N*4
    float* pkY    = pkX + (size_t)B * N * 4;        // B*M*4
    float* rowMin = pkY + (size_t)B * M * 4;        // B*N
    float* colMin = rowMin + (size_t)B * N;         // B*M

    // Pack both clouds into [p0,p1,p2,|p|^2] float4s.
    pack_points<<<(B * N + 255) / 256, 256, 0, stream>>>(x, pkX, B * N);
    pack_points<<<(B * M + 255) / 256, 256, 0, stream>>>(y, pkY, B * M);

    // Pass A: for each x row, min over all y (axis -1 min).
    {
        const int waves  = B * (N >> 4);
        const int blocks = (waves * WAVE) / BLOCK;
        chamfer_min_pass<<<blocks, BLOCK, 0, stream>>>(x, pkY, rowMin, N, M);
    }
    // Pass B: roles swapped -> for each y row, min over all x (axis -2 min).
    {
        const int waves  = B * (M >> 4);
        const int blocks = (waves * WAVE) / BLOCK;
        chamfer_min_pass<<<blocks, BLOCK, 0, stream>>>(y, pkX, colMin, M, N);
    }
    // Final deterministic reduction into the scalar output.
    chamfer_reduce<<<1, 256, 0, stream>>>(rowMin, B * (N + M), out);
}